// BlockCirculantLinear_28269474742318
// MI455X (gfx1250) — compile-verified
//
#include <hip/hip_runtime.h>

// Block-circulant linear: out = (x * D) @ M^T where M[o*128+t, j*128+s] = W[o,j,(s-t) & 127].
// Dense-GEMM formulation with on-the-fly circulant expansion from LDS.
// f32 WMMA (V_WMMA_F32_16X16X4_F32) -> exact w.r.t. the f32 reference.

typedef __attribute__((ext_vector_type(2))) float v2f;
typedef __attribute__((ext_vector_type(8))) float v8f;

constexpr int BATCH = 4096;
constexpr int D_IN  = 8192;
constexpr int D_OUT = 8192;
constexpr int BLK   = 128;            // circulant block size
constexpr int K_IN  = D_IN / BLK;     // 64
constexpr int K_OUT = D_OUT / BLK;    // 64

constexpr int TILE_M = 64;            // batch rows per workgroup
constexpr int PAD    = 4;             // LDS pad: column reads hit distinct banks

__global__ __launch_bounds__(256)
void bcirc_gemm_wmma_f32(const float* __restrict__ x,
                         const float* __restrict__ W,
                         const float* __restrict__ D,
                         float* __restrict__ out) {
  __shared__ float xs[TILE_M][BLK + PAD];  // sign-flipped activation tile
  __shared__ float wd[2 * BLK];            // doubled circulant first row (wrap-free idx,idx+1)

  const int tid  = threadIdx.x;
  const int lane = tid & 31;          // wave32
  const int wave = tid >> 5;          // 0..7
  const int wm   = wave & 3;          // wave row group  (4 x 16 rows)
  const int wn   = wave >> 2;         // wave col group  (2 x 64 cols)
  const int l16  = lane & 15;
  const int lh   = lane >> 4;         // 0/1 (half-wave)

  const int oblk = blockIdx.x;              // output block column, 0..63
  const int m0   = blockIdx.y * TILE_M;     // batch row base
  const int n0   = oblk * BLK;              // output col base

  v8f acc0 = {}, acc1 = {}, acc2 = {}, acc3 = {};

  for (int j = 0; j < K_IN; ++j) {
    // ---- stage circulant first row, doubled ----
    if (tid < BLK) {
      float wv = W[((size_t)oblk * K_IN + j) * BLK + tid];
      wd[tid]       = wv;
      wd[tid + BLK] = wv;
    }
    // ---- stage x tile with diagonal signs applied ----
    {
      const int c4 = tid & 31;        // float4 column 0..31
      const int r0 = tid >> 5;        // 0..7
      const float4* D4 = (const float4*)(D + j * BLK);
      const float4 dv = D4[c4];
#pragma unroll
      for (int rr = 0; rr < TILE_M / 8; ++rr) {
        const int row = rr * 8 + r0;
        const float4* x4 = (const float4*)(x + (size_t)(m0 + row) * D_IN + j * BLK);
        const float4 v = x4[c4];
        xs[row][c4 * 4 + 0] = v.x * dv.x;
        xs[row][c4 * 4 + 1] = v.y * dv.y;
        xs[row][c4 * 4 + 2] = v.z * dv.z;
        xs[row][c4 * 4 + 3] = v.w * dv.w;
        if (j + 1 < K_IN)   // pull next K-block toward L2/L0 (global_prefetch_b8)
          __builtin_prefetch(x + (size_t)(m0 + row) * D_IN + (j + 1) * BLK + c4 * 4, 0, 1);
      }
    }
    __syncthreads();

    // ---- 128-deep K chunk via 32 x V_WMMA_F32_16X16X4_F32 per sub-tile ----
    const int arow = wm * 16 + l16;
    const int nn   = wn * 64 + l16;   // + q*16 below
#pragma unroll
    for (int kk = 0; kk < BLK / 4; ++kk) {
      const int k0 = kk * 4 + 2 * lh; // per-lane K slots (A layout: lanes0-15 K{0,1}, 16-31 K{2,3})
      v2f a;
      a.x = xs[arow][k0];
      a.y = xs[arow][k0 + 1];
#pragma unroll
      for (int q = 0; q < 4; ++q) {
        // B[s][n] = wd[(s - n) & 127] : circulant expansion straight from LDS
        const int idx = (k0 + 2 * BLK - (nn + q * 16)) & (BLK - 1);
        v2f b;
        b.x = wd[idx];
        b.y = wd[idx + 1];
        v8f& acc = (q == 0) ? acc0 : (q == 1) ? acc1 : (q == 2) ? acc2 : acc3;
        acc = __builtin_amdgcn_wmma_f32_16x16x4_f32(
            /*neg_a=*/false, a, /*neg_b=*/false, b,
            /*c_mod=*/(short)0, acc, /*reuse_a=*/false, /*reuse_b=*/false);
      }
    }
    __syncthreads();
  }

  // ---- write back (C/D layout: VGPR r -> M = r + 8*half, lane%16 -> N) ----
#pragma unroll
  for (int q = 0; q < 4; ++q) {
    const v8f acc = (q == 0) ? acc0 : (q == 1) ? acc1 : (q == 2) ? acc2 : acc3;
    const int n = n0 + wn * 64 + q * 16 + l16;
#pragma unroll
    for (int r = 0; r < 8; ++r) {
      const int m = m0 + wm * 16 + 8 * lh + r;
      out[(size_t)m * D_OUT + n] = acc[r];
    }
  }
}

extern "C" void kernel_launch(void* const* d_in, const int* in_sizes, int n_in,
                              void* d_out, int out_size, void* d_ws, size_t ws_size,
                              hipStream_t stream) {
  const float* x = (const float*)d_in[0];   // (4096, 8192)
  const float* W = (const float*)d_in[1];   // (64, 64, 128)
  const float* D = (const float*)d_in[2];   // (8192,)
  float* out = (float*)d_out;               // (4096, 8192)

  dim3 grid(K_OUT, BATCH / TILE_M);         // 64 x 64 workgroups
  dim3 block(256);                          // 8 waves (wave32)
  bcirc_gemm_wmma_f32<<<grid, block, 0, stream>>>(x, W, D, out);
}